// TuplemaxLoss_59090160058685
// MI455X (gfx1250) — compile-verified
//
#include <hip/hip_runtime.h>
#include <hip/hip_bf16.h>
#include <stdint.h>

// ---------------------------------------------------------------------------
// TuplemaxLoss for MI455X (gfx1250, wave32).
//   Kernel 1: TDM (tensor_load_to_lds) double-buffered streaming of 8-row
//             tiles into LDS; one wave per row computes sum_k logaddexp(x_k,t)
//             with v_exp_f32/v_log_f32; per-row result -> d_ws (deterministic).
//   Kernel 2: single-block fixed-order f64 reduction -> mean -> d_out[0].
// ---------------------------------------------------------------------------

#define ROWS_PER_TILE   8
#define BLOCK_THREADS   256     // 8 wave32 waves
#define TILES_PER_BLOCK 4
#define MAX_M           1024

typedef __attribute__((ext_vector_type(4))) unsigned int tdm_v4u;
typedef __attribute__((ext_vector_type(8))) int          tdm_v8i;
typedef __attribute__((ext_vector_type(4))) int          tdm_v4i;

#if __has_builtin(__builtin_amdgcn_tensor_load_to_lds) && __has_builtin(__builtin_amdgcn_s_wait_tensorcnt)
#define USE_TDM 1
#else
#define USE_TDM 0
#endif

// pairwise term: logaddexp(x, t) = max(x,t) + ln2 * log2(1 + exp2(-|x-t|*log2e))
__device__ __forceinline__ float logaddexp_term(float x, float t) {
    float mx = fmaxf(x, t);
    float ad = fabsf(x - t);
    float z  = __builtin_amdgcn_exp2f(ad * -1.4426950408889634f);  // v_exp_f32
    float l  = __builtin_amdgcn_logf(1.0f + z);                    // v_log_f32 (log2)
    return fmaf(l, 0.69314718055994530942f, mx);
}

#if USE_TDM
// Build a 2D tile descriptor (tile = M x ROWS_PER_TILE fp32, dense into LDS)
// and issue the TDM load. gaddr points at the TILE start; lds_off is the LDS
// byte address of the destination buffer.
__device__ __forceinline__ void tdm_issue(unsigned long long gaddr,
                                          unsigned lds_off, int M, int N) {
    tdm_v4u g0;
    g0[0] = 1u;                                         // count=1 (valid), user mode
    g0[1] = lds_off;                                    // lds_addr [63:32]
    g0[2] = (unsigned)(gaddr & 0xffffffffull);          // global_addr[31:0]
    g0[3] = ((unsigned)(gaddr >> 32) & 0x01ffffffu)     // global_addr[56:32]
            | 0x80000000u;                              // type=2 ("image")
    tdm_v8i g1;
    g1[0] = 0x00020000;                                 // data_size=2 (4 bytes); no mask/pad/iter
    g1[1] = (int)(((unsigned)M & 0xffffu) << 16);       // tensor_dim0[15:0] @ bits[63:48]
    g1[2] = (int)((((unsigned)M >> 16) & 0xffffu)       // tensor_dim0[31:16]
            | (((unsigned)N & 0xffffu) << 16));         // tensor_dim1[15:0]
    g1[3] = (int)((((unsigned)N >> 16) & 0xffffu)       // tensor_dim1[31:16]
            | (((unsigned)M & 0xffffu) << 16));         // tile_dim0 = M
    g1[4] = ROWS_PER_TILE;                              // tile_dim1 = 8, tile_dim2 = 0
    g1[5] = M;                                          // tensor_dim0_stride[31:0]
    g1[6] = 0;                                          // stride0 hi, dim1_stride lo
    g1[7] = 0;                                          // dim1_stride hi
    tdm_v4i gz4 = {0, 0, 0, 0};                         // group2 (unused: 2D tile)
    tdm_v8i gz8 = {0, 0, 0, 0, 0, 0, 0, 0};             // group3/extra words (unused)
    // clang-23 / amdgpu-toolchain 6-arg form:
    //   (v4u g0, v8i g1, v4i g2, v4i g3, v8i extra, i32 cpol)
    __builtin_amdgcn_tensor_load_to_lds(g0, g1, gz4, gz4, gz8, 0);
}
#endif

__global__ __launch_bounds__(BLOCK_THREADS)
void tuplemax_rows_kernel(const float* __restrict__ logits,
                          const long long* __restrict__ true_idx,
                          float* __restrict__ rowout,
                          int N, int M, int tilesTotal) {
    const int lane = threadIdx.x & 31;
    const int wave = threadIdx.x >> 5;

#if USE_TDM
    __shared__ float buf[2][ROWS_PER_TILE * MAX_M];     // 2 x 32 KB

    const int tileBase = blockIdx.x * TILES_PER_BLOCK;
    int nT = tilesTotal - tileBase;
    if (nT <= 0) return;
    if (nT > TILES_PER_BLOCK) nT = TILES_PER_BLOCK;

    if (threadIdx.x == 0) {   // wave0 drives the DMA pipeline (TDM ignores EXEC)
        tdm_issue((unsigned long long)(uintptr_t)(logits + (size_t)tileBase * ROWS_PER_TILE * M),
                  (unsigned)(uintptr_t)(&buf[0][0]), M, N);
    }

    for (int j = 0; j < nT; ++j) {
        if (threadIdx.x == 0) {
            if (j + 1 < nT) {
                tdm_issue((unsigned long long)(uintptr_t)(logits + (size_t)(tileBase + j + 1) * ROWS_PER_TILE * M),
                          (unsigned)(uintptr_t)(&buf[(j + 1) & 1][0]), M, N);
                __builtin_amdgcn_s_wait_tensorcnt(1);   // tile j has landed
            } else {
                __builtin_amdgcn_s_wait_tensorcnt(0);   // last tile
            }
        }
        __syncthreads();                                 // tile j visible to all waves

        const int row = (tileBase + j) * ROWS_PER_TILE + wave;
        if (row < N) {
            const float* r = &buf[j & 1][wave * M];
            const float  t = r[(int)true_idx[row]];      // LDS broadcast read

            float s0 = 0.f, s1 = 0.f, s2 = 0.f, s3 = 0.f;
            const float4* rv = (const float4*)r;
            const int iters = M >> 7;                    // M / (32 lanes * 4)
            for (int it = 0; it < iters; ++it) {
                float4 v = rv[it * 32 + lane];
                s0 += logaddexp_term(v.x, t);
                s1 += logaddexp_term(v.y, t);
                s2 += logaddexp_term(v.z, t);
                s3 += logaddexp_term(v.w, t);
            }
            float s = (s0 + s1) + (s2 + s3);
            #pragma unroll
            for (int off = 16; off > 0; off >>= 1)
                s += __shfl_xor(s, off, 32);             // wave32 reduction
            if (lane == 0) {
                float logsum = s - (t + 0.69314718055994530942f);
                rowout[row]  = logsum / (float)(M - 1) - t;
            }
        }
        __syncthreads();                                 // buffer reusable for next DMA
    }
#else
    // Fallback: direct global float4 streaming, one wave per row, grid-stride.
    const int wavesPerBlock = BLOCK_THREADS / 32;
    const int totalWaves = (int)gridDim.x * wavesPerBlock;
    (void)tilesTotal;
    for (int row = (int)blockIdx.x * wavesPerBlock + wave; row < N; row += totalWaves) {
        const float* r = logits + (size_t)row * M;
        const float  t = r[(int)true_idx[row]];
        float s0 = 0.f, s1 = 0.f, s2 = 0.f, s3 = 0.f;
        const float4* rv = (const float4*)r;
        const int iters = M >> 7;
        for (int it = 0; it < iters; ++it) {
            float4 v = rv[it * 32 + lane];
            s0 += logaddexp_term(v.x, t);
            s1 += logaddexp_term(v.y, t);
            s2 += logaddexp_term(v.z, t);
            s3 += logaddexp_term(v.w, t);
        }
        float s = (s0 + s1) + (s2 + s3);
        #pragma unroll
        for (int off = 16; off > 0; off >>= 1)
            s += __shfl_xor(s, off, 32);
        if (lane == 0) {
            float logsum = s - (t + 0.69314718055994530942f);
            rowout[row]  = logsum / (float)(M - 1) - t;
        }
    }
#endif
}

// Deterministic fixed-order final reduction (f64 accumulation, tiny work).
__global__ __launch_bounds__(256)
void tuplemax_reduce_kernel(const float* __restrict__ rowvals,
                            float* __restrict__ out, int N) {
    __shared__ double sh[256];
    double s = 0.0;
    for (int i = threadIdx.x; i < N; i += 256) s += (double)rowvals[i];
    sh[threadIdx.x] = s;
    __syncthreads();
    for (int off = 128; off > 0; off >>= 1) {
        if ((int)threadIdx.x < off) sh[threadIdx.x] += sh[threadIdx.x + off];
        __syncthreads();
    }
    if (threadIdx.x == 0) out[0] = (float)(sh[0] / (double)N);
}

extern "C" void kernel_launch(void* const* d_in, const int* in_sizes, int n_in,
                              void* d_out, int out_size, void* d_ws, size_t ws_size,
                              hipStream_t stream) {
    const float*     logits   = (const float*)d_in[0];
    const long long* true_idx = (const long long*)d_in[1];
    float*           out      = (float*)d_out;
    float*           rowvals  = (float*)d_ws;          // N floats of scratch

    const int N = in_sizes[1];                         // 65536
    const int M = in_sizes[0] / N;                     // 1024

    const int tilesTotal = (N + ROWS_PER_TILE - 1) / ROWS_PER_TILE;
    const int grid       = (tilesTotal + TILES_PER_BLOCK - 1) / TILES_PER_BLOCK;

    tuplemax_rows_kernel<<<grid, BLOCK_THREADS, 0, stream>>>(
        logits, true_idx, rowvals, N, M, tilesTotal);
    tuplemax_reduce_kernel<<<1, 256, 0, stream>>>(rowvals, out, N);
}